// ConvSV_NB_24816321036386
// MI455X (gfx1250) — compile-verified
//
#include <hip/hip_runtime.h>

typedef __attribute__((ext_vector_type(2))) float v2f;
typedef __attribute__((ext_vector_type(8))) float v8f;

#define MM 48   // neighbors (K dim)
#define GG 16   // shifts
#define AA 64   // channels
#define HH 16   // comb_v
#define WPB 4   // waves per block (one atom per wave)

// Per atom: D[64 x 64] = a^T[64 x 48] * [gs | gv][48 x 64] via v_wmma_f32_16x16x4_f32.
// Columns 0-15  -> gs   (scalar branch, stored directly)
// Columns 16-63 -> gv   (T matrix, staged in LDS for the agh contraction epilogue)
__global__ __launch_bounds__(WPB * 32) void conv_sv_wmma(
    const float* __restrict__ a, const float* __restrict__ gs,
    const float* __restrict__ gv, const float* __restrict__ agh,
    float* __restrict__ out, int n_atoms)
{
    __shared__ float t_lds[WPB][AA * GG * 3];  // 4 x 12KB = 48KB

    const int lane = threadIdx.x & 31;
    const int wave = threadIdx.x >> 5;
    const int n = blockIdx.x * WPB + wave;
    if (n >= n_atoms) return;  // uniform per wave -> EXEC stays all-ones

    const int hi = lane >> 4;   // 0: lanes 0-15, 1: lanes 16-31
    const int lo = lane & 15;

    const float* __restrict__ An = a  + (size_t)n * MM * AA;        // [m][a]
    const float* __restrict__ Sn = gs + (size_t)n * MM * GG;        // [m][g]
    const float* __restrict__ Vn = gv + (size_t)n * MM * GG * 3;    // [m][g*3+d]

    v8f acc[4][4];
    #pragma unroll
    for (int t = 0; t < 4; ++t)
        #pragma unroll
        for (int c = 0; c < 4; ++c)
            #pragma unroll
            for (int r = 0; r < 8; ++r)
                acc[t][c][r] = 0.0f;

    // K loop: 12 steps of K=4. Per ISA 32-bit A layout (16x4):
    //   lanes 0-15 : VGPR0=K0, VGPR1=K1 ; lanes 16-31 : VGPR0=K2, VGPR1=K3
    // B (4x16) row-striped: VGPR0 = row K0/K2 across lanes, VGPR1 = row K1/K3.
    #pragma unroll
    for (int k = 0; k < MM / 4; ++k) {
        const int m0 = 4 * k + 2 * hi;

        v2f af[4], bf[4];
        #pragma unroll
        for (int t = 0; t < 4; ++t) {             // A fragments: a^T tiles
            af[t].x = An[(m0    ) * AA + t * 16 + lo];
            af[t].y = An[(m0 + 1) * AA + t * 16 + lo];
        }
        bf[0].x = Sn[(m0    ) * GG + lo];         // B tile 0: gs
        bf[0].y = Sn[(m0 + 1) * GG + lo];
        #pragma unroll
        for (int c = 1; c < 4; ++c) {             // B tiles 1-3: gv columns
            const int col = (c - 1) * 16 + lo;
            bf[c].x = Vn[(m0    ) * 48 + col];
            bf[c].y = Vn[(m0 + 1) * 48 + col];
        }

        #pragma unroll
        for (int t = 0; t < 4; ++t)
            #pragma unroll
            for (int c = 0; c < 4; ++c)
                acc[t][c] = __builtin_amdgcn_wmma_f32_16x16x4_f32(
                    false, af[t], false, bf[c], (short)0, acc[t][c], false, false);
    }

    // ---- scalar branch: out_s[n, a*16 + g] -------------------------------
    // C/D layout: VGPR r, lanes 0-15 -> M=r, lanes 16-31 -> M=r+8, N=lane&15
    float* __restrict__ outS = out + (size_t)n * (AA * GG + AA * HH);
    #pragma unroll
    for (int t = 0; t < 4; ++t)
        #pragma unroll
        for (int r = 0; r < 8; ++r) {
            const int ai = t * 16 + r + 8 * hi;
            outS[ai * GG + lo] = acc[t][0][r];
        }

    // ---- stage T[a][g*3+d] into this wave's private LDS region ----------
    float* __restrict__ Tw = &t_lds[wave][0];
    #pragma unroll
    for (int t = 0; t < 4; ++t)
        #pragma unroll
        for (int c = 1; c < 4; ++c)
            #pragma unroll
            for (int r = 0; r < 8; ++r) {
                const int ai  = t * 16 + r + 8 * hi;
                const int col = (c - 1) * 16 + lo;
                Tw[ai * 48 + col] = acc[t][c][r];
            }

    // wave-private LDS region: a dscnt drain is sufficient (no block barrier)
    asm volatile("s_wait_dscnt 0" ::: "memory");

    // ---- vector branch epilogue: out_v[n, a*16 + h] ----------------------
    float* __restrict__ outV = outS + AA * GG;
    #pragma unroll
    for (int half = 0; half < 2; ++half) {
        const int ai = half * 32 + lane;
        const float* __restrict__ W  = agh + (size_t)ai * GG * HH;  // [g][h]
        const float* __restrict__ Ta = &t_lds[wave][ai * 48];

        float acc0[HH], acc1[HH], acc2[HH];
        #pragma unroll
        for (int h = 0; h < HH; ++h) { acc0[h] = 0.f; acc1[h] = 0.f; acc2[h] = 0.f; }

        for (int g = 0; g < GG; ++g) {
            const float td0 = Ta[g * 3 + 0];
            const float td1 = Ta[g * 3 + 1];
            const float td2 = Ta[g * 3 + 2];
            #pragma unroll
            for (int h = 0; h < HH; ++h) {        // h-contiguous -> b128 loads
                const float w = W[g * HH + h];
                acc0[h] = fmaf(w, td0, acc0[h]);
                acc1[h] = fmaf(w, td1, acc1[h]);
                acc2[h] = fmaf(w, td2, acc2[h]);
            }
        }
        #pragma unroll
        for (int h = 0; h < HH; ++h)
            outV[ai * HH + h] =
                acc0[h] * acc0[h] + acc1[h] * acc1[h] + acc2[h] * acc2[h];
    }
}

extern "C" void kernel_launch(void* const* d_in, const int* in_sizes, int n_in,
                              void* d_out, int out_size, void* d_ws, size_t ws_size,
                              hipStream_t stream) {
    const float* a   = (const float*)d_in[0];   // [N, 48, 64]
    const float* gs  = (const float*)d_in[1];   // [N, 48, 16]
    const float* gv  = (const float*)d_in[2];   // [N, 48, 16, 3]
    const float* agh = (const float*)d_in[3];   // [64, 16, 16]
    float* out = (float*)d_out;                 // [N, 2048]

    const int n_atoms = in_sizes[0] / (MM * AA);
    const int blocks = (n_atoms + WPB - 1) / WPB;
    conv_sv_wmma<<<blocks, WPB * 32, 0, stream>>>(a, gs, gv, agh, out, n_atoms);
}